// SpatialNet_66529043415173
// MI455X (gfx1250) — compile-verified
//
#include <hip/hip_runtime.h>
#include <math.h>

typedef __attribute__((ext_vector_type(2))) float v2f;
typedef __attribute__((ext_vector_type(8))) float v8f;

constexpr int JN   = 25;     // joints
constexpr int JP   = 32;     // padded joints
constexpr int TT   = 512;    // time steps
constexpr int BSN  = 32;     // batch
constexpr int D2C  = 128;    // inp channels
constexpr int DIMC = 256;    // output channels
constexpr int NBT  = BSN * TT;            // 16384 (b,t) pairs
constexpr int BSTR = 3 * JN * TT;         // 38400: batch stride of Angle/dif
constexpr float EPSV = 1e-5f;

// ---- workspace layout (floats) ----
constexpr size_t OFF_INP  = 0;
constexpr size_t N_INP    = (size_t)NBT * D2C * JN;        // 52,428,800
constexpr size_t OFF_BNS  = OFF_INP + N_INP;               // 150 (embed BN scale)
constexpr size_t OFF_BNB  = OFF_BNS + 150;                 // 150 (embed BN shift)
constexpr size_t OFF_SPA  = OFF_BNB + 150;                 // 64*25
constexpr size_t OFF_HMAX = OFF_SPA + 64 * JN;             // NBT*256
constexpr size_t OFF_HMIN = OFF_HMAX + (size_t)NBT * DIMC;
constexpr size_t OFF_PSUM = OFF_HMIN + (size_t)NBT * DIMC;
constexpr size_t OFF_PSQ  = OFF_PSUM + (size_t)NBT * DIMC;
constexpr size_t OFF_GM   = OFF_PSQ  + (size_t)NBT * DIMC; // 256
constexpr size_t OFF_GI   = OFF_GM + DIMC;                 // 256

// =====================================================================
// Kernel 1: training-mode BN stats for the two embed inputs (150 chans)
// =====================================================================
__global__ void k_bn_embed_stats(const float* __restrict__ Angle, const float* __restrict__ dif,
                                 const float* __restrict__ jg, const float* __restrict__ jb,
                                 const float* __restrict__ dg, const float* __restrict__ db,
                                 float* __restrict__ bnscale, float* __restrict__ bnshift) {
  const int ch  = blockIdx.x;          // 0..149
  const int tid = threadIdx.x;
  const float* X; float gamma, beta; int cj;
  if (ch < 75) { X = Angle; cj = ch;      gamma = jg[cj]; beta = jb[cj]; }
  else         { X = dif;   cj = ch - 75; gamma = dg[cj]; beta = db[cj]; }
  const float* base = X + (size_t)cj * TT;
  float s = 0.f, s2 = 0.f;
  for (int n = tid; n < NBT; n += 256) {
    int b = n >> 9, t = n & 511;
    float v = base[(size_t)b * BSTR + t];
    s += v; s2 += v * v;
  }
  __shared__ float rs[256], rq[256];
  rs[tid] = s; rq[tid] = s2; __syncthreads();
  for (int off = 128; off > 0; off >>= 1) {
    if (tid < off) { rs[tid] += rs[tid + off]; rq[tid] += rq[tid + off]; }
    __syncthreads();
  }
  if (tid == 0) {
    float m   = rs[0] * (1.0f / 16384.0f);
    float var = rq[0] * (1.0f / 16384.0f) - m * m;
    float inv = 1.0f / sqrtf(var + EPSV);
    float scl = gamma * inv;
    bnscale[ch] = scl;
    bnshift[ch] = beta - m * scl;
  }
}

// =====================================================================
// Kernel 2: spa embedding (tiny): spa_emb[64][25]
// =====================================================================
__global__ void k_spa_embed(const float* __restrict__ spa,
                            const float* __restrict__ w1, const float* __restrict__ b1,
                            const float* __restrict__ w2, const float* __restrict__ b2,
                            float* __restrict__ spa_emb) {
  const int j = threadIdx.x;
  if (j >= JN) return;
  float h1[64];
  for (int o = 0; o < 64; ++o) {
    float v = b1[o];
    for (int c = 0; c < JN; ++c) v += w1[o * JN + c] * spa[c * JN + j];
    h1[o] = fmaxf(v, 0.f);
  }
  for (int o2 = 0; o2 < 64; ++o2) {
    float v = b2[o2];
    for (int o = 0; o < 64; ++o) v += w2[o2 * 64 + o] * h1[o];
    spa_emb[o2 * JN + j] = fmaxf(v, 0.f);
  }
}

// =====================================================================
// Kernel 3 (WMMA): build inp[b][t][c][j].
// One workgroup = 32 consecutive t for a fixed (b,j). Layer-1 (K=3) is
// VALU into LDS; layer-2 (64x64 GEMM over 32 columns) is WMMA f32
// 16x16x4. Wave w computes the SAME output tile for the Angle path
// (job w) and dif path (job w+8), so relu(a)+relu(d) combines in regs.
// =====================================================================
__global__ void __launch_bounds__(256) k_embed_gemm(
    const float* __restrict__ Angle, const float* __restrict__ dif,
    const float* __restrict__ jw1, const float* __restrict__ jb1,
    const float* __restrict__ jw2, const float* __restrict__ jb2,
    const float* __restrict__ dw1, const float* __restrict__ db1,
    const float* __restrict__ dw2, const float* __restrict__ db2,
    const float* __restrict__ bnscale, const float* __restrict__ bnshift,
    const float* __restrict__ spa_emb,
    float* __restrict__ inp) {
  __shared__ float xn[2 * 3 * 32];       // normalized inputs [path][c][p]
  __shared__ float h1buf[2][64 * 32];    // relu(layer1) [path][o][p]

  const int tid  = threadIdx.x;
  const int wave = tid >> 5;
  const int lane = tid & 31;
  const int hi   = lane >> 4;
  const int lo   = lane & 15;

  const int blk = blockIdx.x;            // 32*25*16 = 12800 blocks
  const int b   = blk / 400;
  const int rem = blk % 400;
  const int j   = rem / 16;
  const int t0  = (rem % 16) * 32;

  // ---- Phase 0: load + normalize 3 channels x 32 t for both tensors
  if (tid < 192) {
    const int s  = tid / 96;
    const int r3 = tid % 96;
    const int c  = r3 / 32;
    const int p  = r3 % 32;
    const int cj = c * JN + j;
    const float* X = s ? dif : Angle;
    const float raw = X[(size_t)b * BSTR + (size_t)cj * TT + t0 + p];
    xn[tid] = raw * bnscale[cj + s * 75] + bnshift[cj + s * 75];
  }
  __syncthreads();

  // ---- Phase 1: layer-1 (K=3) into LDS: h1[s][o][p]
  for (int i = tid; i < 4096; i += 256) {
    const int s  = i >> 11;
    const int r3 = i & 2047;
    const int o  = r3 >> 5;
    const int p  = r3 & 31;
    const float* w1 = s ? dw1 : jw1;
    const float* b1 = s ? db1 : jb1;
    const float* x  = xn + s * 96;
    float v = b1[o] + w1[o*3]*x[p] + w1[o*3+1]*x[32+p] + w1[o*3+2]*x[64+p];
    h1buf[s][o * 32 + p] = fmaxf(v, 0.f);
  }
  __syncthreads();

  // ---- Phase 2: layer-2 WMMA GEMMs (a-path and d-path, same tile pos)
  const int mt = wave >> 1;              // 0..3 (output-channel tile)
  const int nt = wave & 1;               // 0..1 (column tile)
  const int rowA = mt * 16 + lo;
  const int coln = nt * 16 + lo;

  v8f accA, accD;
  for (int r2 = 0; r2 < 8; ++r2) {
    accA[r2] = jb2[mt * 16 + hi * 8 + r2];
    accD[r2] = db2[mt * 16 + hi * 8 + r2];
  }
  for (int kk = 0; kk < 16; ++kk) {
    const int k0 = kk * 4 + hi * 2;
    v2f a, bf;
    const float2 wv = *(const float2*)(jw2 + rowA * 64 + k0);
    a[0] = wv.x; a[1] = wv.y;
    bf[0] = h1buf[0][k0 * 32 + coln];
    bf[1] = h1buf[0][(k0 + 1) * 32 + coln];
    accA = __builtin_amdgcn_wmma_f32_16x16x4_f32(false, a, false, bf, (short)0, accA, false, false);
  }
  for (int kk = 0; kk < 16; ++kk) {
    const int k0 = kk * 4 + hi * 2;
    v2f a, bf;
    const float2 wv = *(const float2*)(dw2 + rowA * 64 + k0);
    a[0] = wv.x; a[1] = wv.y;
    bf[0] = h1buf[1][k0 * 32 + coln];
    bf[1] = h1buf[1][(k0 + 1) * 32 + coln];
    accD = __builtin_amdgcn_wmma_f32_16x16x4_f32(false, a, false, bf, (short)0, accD, false, false);
  }

  // ---- Phase 3: relu(a)+relu(d) -> inp, plus broadcast spa channels
  for (int r2 = 0; r2 < 8; ++r2) {
    const int c = mt * 16 + hi * 8 + r2;           // output channel 0..63
    const int p = nt * 16 + lo;                    // column 0..31
    const int t = t0 + p;
    const float v = fmaxf(accA[r2], 0.f) + fmaxf(accD[r2], 0.f);
    inp[((size_t)(b * TT + t) * D2C + c) * JN + j] = v;
  }
  for (int i = tid; i < 2048; i += 256) {
    const int c = i >> 5;
    const int p = i & 31;
    inp[((size_t)(b * TT + t0 + p) * D2C + 64 + c) * JN + j] = spa_emb[c * JN + j];
  }
}

// =====================================================================
// Kernel 4: fused attention per (b,t): WMMA f32 16x16x4 everywhere
// =====================================================================
__global__ void __launch_bounds__(256) k_attn(
    const float* __restrict__ inp,
    const float* __restrict__ g1w, const float* __restrict__ g1b,
    const float* __restrict__ g2w, const float* __restrict__ g2b,
    const float* __restrict__ gw,  const float* __restrict__ gw1, const float* __restrict__ gw1b,
    float* __restrict__ hmax, float* __restrict__ hmin,
    float* __restrict__ psum, float* __restrict__ psq) {
  extern __shared__ float smem[];
  float* xs = smem;          // [128][32] x tile (j-padded with zeros)
  float* qs = smem + 4096;   // [64][32]  q channel-chunk
  float* ks = smem + 6144;   // [64][32]  k channel-chunk
  float* sc = smem + 8192;   // [32][32]  scores / softmax(g)
  float* xm = smem + 9216;   // [128][32] g-mixed x

  const int tid  = threadIdx.x;
  const int wave = tid >> 5;
  const int lane = tid & 31;
  const int hi   = lane >> 4;   // 0/1: which half of the wave
  const int lo   = lane & 15;
  const int nbt  = blockIdx.x;  // b*512 + t
  const float* xg = inp + (size_t)nbt * (D2C * JN);

  // ---- Phase 0: stage x into LDS (zero the 7 pad columns)
  for (int i = tid; i < 128 * 7; i += 256) {
    int c = i / 7, p = i % 7;
    xs[c * JP + JN + p] = 0.f;
  }
  for (int i = tid; i < D2C * JN; i += 256) {
    int c = i / JN, j = i % JN;
    xs[c * JP + j] = xg[i];
  }
  __syncthreads();

  // ---- Phases 1+2: q/k in 64-channel chunks; scores S = q^T k accumulated
  v8f sacc = {};
  const int smt = wave >> 1, snt = wave & 1;  // score tile owned by waves 0..3
  for (int chunk = 0; chunk < 4; ++chunk) {
    for (int job = wave; job < 16; job += 8) {      // 4 mt x 2 nt x {q,k}
      const int mt  = job >> 2;
      const int nt  = (job >> 1) & 1;
      const int isK = job & 1;
      const float* W  = isK ? g2w : g1w;
      const float* Bv = isK ? g2b : g1b;
      float* dst = isK ? ks : qs;
      const int m0 = chunk * 64 + mt * 16;
      v8f acc;
      for (int r2 = 0; r2 < 8; ++r2) acc[r2] = Bv[m0 + hi * 8 + r2];
      const int rowA = m0 + lo;
      const int coln = nt * 16 + lo;
      for (int kk = 0; kk < 32; ++kk) {
        const int k0 = kk * 4 + hi * 2;
        v2f a, bf;
        const float2 wv = *(const float2*)(W + rowA * 128 + k0);
        a[0] = wv.x; a[1] = wv.y;
        bf[0] = xs[k0 * JP + coln];
        bf[1] = xs[(k0 + 1) * JP + coln];
        acc = __builtin_amdgcn_wmma_f32_16x16x4_f32(false, a, false, bf, (short)0, acc, false, false);
      }
      const int mloc = mt * 16 + hi * 8;
      for (int r2 = 0; r2 < 8; ++r2) dst[(mloc + r2) * JP + nt * 16 + lo] = acc[r2];
    }
    __syncthreads();
    if (wave < 4) {                                  // S += Qc^T Kc (K=64)
      const int rowA = smt * 16 + lo;                // i
      const int coln = snt * 16 + lo;                // j
      for (int kk = 0; kk < 16; ++kk) {
        const int k0 = kk * 4 + hi * 2;              // local channel
        v2f a, bf;
        a[0] = qs[k0 * JP + rowA];
        a[1] = qs[(k0 + 1) * JP + rowA];
        bf[0] = ks[k0 * JP + coln];
        bf[1] = ks[(k0 + 1) * JP + coln];
        sacc = __builtin_amdgcn_wmma_f32_16x16x4_f32(false, a, false, bf, (short)0, sacc, false, false);
      }
    }
    __syncthreads();
  }
  if (wave < 4) {
    const int mloc = smt * 16 + hi * 8;
    for (int r2 = 0; r2 < 8; ++r2) sc[(mloc + r2) * JP + snt * 16 + lo] = sacc[r2];
  }
  __syncthreads();

  // ---- Phase 3: row softmax (wave 0, lane = row); zero padded rows/cols
  if (wave == 0) {
    const int row = lane;
    if (row < JN) {
      float m = -3.402823466e38f;
      for (int j2 = 0; j2 < JN; ++j2) m = fmaxf(m, sc[row * JP + j2]);
      float e[JN]; float s = 0.f;
      for (int j2 = 0; j2 < JN; ++j2) { e[j2] = expf(sc[row * JP + j2] - m); s += e[j2]; }
      const float invs = 1.0f / s;
      for (int j2 = 0; j2 < JN; ++j2) sc[row * JP + j2] = e[j2] * invs;
      for (int j2 = JN; j2 < JP; ++j2) sc[row * JP + j2] = 0.f;
    } else {
      for (int j2 = 0; j2 < JP; ++j2) sc[row * JP + j2] = 0.f;
    }
  }
  __syncthreads();

  // ---- Phase 4: xm[c][i] = sum_j x[c][j] * g[i][j]  (M=128,N=32,K=32)
  for (int job = wave; job < 16; job += 8) {
    const int mt = job >> 1, nt = job & 1;
    v8f acc = {};
    const int rowA = mt * 16 + lo;   // c
    const int coln = nt * 16 + lo;   // i
    for (int kk = 0; kk < 8; ++kk) {
      const int k0 = kk * 4 + hi * 2;  // j
      v2f a, bf;
      a[0] = xs[rowA * JP + k0];
      a[1] = xs[rowA * JP + k0 + 1];
      bf[0] = sc[coln * JP + k0];      // B[k=j][n=i] = g[i][j]
      bf[1] = sc[coln * JP + k0 + 1];
      acc = __builtin_amdgcn_wmma_f32_16x16x4_f32(false, a, false, bf, (short)0, acc, false, false);
    }
    const int mloc = mt * 16 + hi * 8;
    for (int r2 = 0; r2 < 8; ++r2) xm[(mloc + r2) * JP + nt * 16 + lo] = acc[r2];
  }
  __syncthreads();

  // ---- Phase 5: H = gw1@x + gw@xm + gw1_b, per-wave tiles mt in {w, w+8}
  v8f hacc[2][2];
  for (int mi = 0; mi < 2; ++mi) {
    const int m0 = (wave + mi * 8) * 16;
    const int rowA = m0 + lo;
    for (int nt = 0; nt < 2; ++nt) {
      const int coln = nt * 16 + lo;
      v8f acc;
      for (int r2 = 0; r2 < 8; ++r2) acc[r2] = gw1b[m0 + hi * 8 + r2];
      for (int kk = 0; kk < 32; ++kk) {
        const int k0 = kk * 4 + hi * 2;
        v2f a, bf;
        const float2 wv = *(const float2*)(gw1 + rowA * 128 + k0);
        a[0] = wv.x; a[1] = wv.y;
        bf[0] = xs[k0 * JP + coln];
        bf[1] = xs[(k0 + 1) * JP + coln];
        acc = __builtin_amdgcn_wmma_f32_16x16x4_f32(false, a, false, bf, (short)0, acc, false, false);
      }
      for (int kk = 0; kk < 32; ++kk) {
        const int k0 = kk * 4 + hi * 2;
        v2f a, bf;
        const float2 wv = *(const float2*)(gw + rowA * 128 + k0);
        a[0] = wv.x; a[1] = wv.y;
        bf[0] = xm[k0 * JP + coln];
        bf[1] = xm[(k0 + 1) * JP + coln];
        acc = __builtin_amdgcn_wmma_f32_16x16x4_f32(false, a, false, bf, (short)0, acc, false, false);
      }
      hacc[mi][nt] = acc;
    }
  }

  // ---- Phase 6: per-channel max/min/sum/sumsq over j<25 via wave32 shuffles
  const bool v1ok = lo < 9;   // nt=1 column j = 16+lo valid iff j < 25
  for (int mi = 0; mi < 2; ++mi) {
    const int mtb = (wave + mi * 8) * 16 + hi * 8;
    for (int r2 = 0; r2 < 8; ++r2) {
      const float a0 = hacc[mi][0][r2];
      const float a1 = hacc[mi][1][r2];
      float mx = fmaxf(a0, v1ok ? a1 : -3.402823466e38f);
      float mn = fminf(a0, v1ok ? a1 :  3.402823466e38f);
      float sm = a0 + (v1ok ? a1 : 0.f);
      float sq = a0 * a0 + (v1ok ? a1 * a1 : 0.f);
      for (int d = 8; d > 0; d >>= 1) {
        mx = fmaxf(mx, __shfl_xor(mx, d, 16));
        mn = fminf(mn, __shfl_xor(mn, d, 16));
        sm += __shfl_xor(sm, d, 16);
        sq += __shfl_xor(sq, d, 16);
      }
      if (lo == 0) {
        const int ch = mtb + r2;
        const size_t o = (size_t)nbt * DIMC + ch;
        hmax[o] = mx; hmin[o] = mn; psum[o] = sm; psq[o] = sq;
      }
    }
  }
}

// =====================================================================
// Kernel 5: global BN stats over the 16384 per-(b,t) partials
// =====================================================================
__global__ void k_bn2_stats(const float* __restrict__ psum, const float* __restrict__ psq,
                            float* __restrict__ gmean, float* __restrict__ ginv) {
  const int c = blockIdx.x;       // 0..255
  const int tid = threadIdx.x;
  float s = 0.f, s2 = 0.f;
  for (int n = tid; n < NBT; n += 256) {
    s  += psum[(size_t)n * DIMC + c];
    s2 += psq [(size_t)n * DIMC + c];
  }
  __shared__ float rs[256], rq[256];
  rs[tid] = s; rq[tid] = s2; __syncthreads();
  for (int off = 128; off > 0; off >>= 1) {
    if (tid < off) { rs[tid] += rs[tid + off]; rq[tid] += rq[tid + off]; }
    __syncthreads();
  }
  if (tid == 0) {
    const float cnt = 1.0f / 409600.0f;   // 32*25*512
    float m   = rs[0] * cnt;
    float var = rq[0] * cnt - m * m;
    gmean[c] = m;
    ginv[c]  = 1.0f / sqrtf(var + EPSV);
  }
}

// =====================================================================
// Kernel 6: finalize  out[b][c][0][t] = relu(bn(h)).max_j  (monotone trick)
// =====================================================================
__global__ void k_finalize(const float* __restrict__ hmax, const float* __restrict__ hmin,
                           const float* __restrict__ gmean, const float* __restrict__ ginv,
                           const float* __restrict__ gg, const float* __restrict__ gb,
                           float* __restrict__ out) {
  const int idx = blockIdx.x * 256 + threadIdx.x;  // (b*256 + c)*512 + t
  const int b   = idx >> 17;
  const int rem = idx & 131071;
  const int c   = rem >> 9;
  const int t   = rem & 511;
  const float a    = gg[c] * ginv[c];
  const float base = gb[c] - gmean[c] * a;
  const size_t o = (size_t)(b * TT + t) * DIMC + c;
  const float h = (a >= 0.f) ? hmax[o] : hmin[o];
  out[idx] = fmaxf(a * h + base, 0.f);
}

// =====================================================================
extern "C" void kernel_launch(void* const* d_in, const int* in_sizes, int n_in,
                              void* d_out, int out_size, void* d_ws, size_t ws_size,
                              hipStream_t stream) {
  (void)in_sizes; (void)n_in; (void)out_size; (void)ws_size;
  const float* Angle   = (const float*)d_in[0];
  const float* dif     = (const float*)d_in[1];
  const float* spa     = (const float*)d_in[2];
  const float* je_bn_g = (const float*)d_in[3];
  const float* je_bn_b = (const float*)d_in[4];
  const float* je_w1   = (const float*)d_in[5];
  const float* je_b1   = (const float*)d_in[6];
  const float* je_w2   = (const float*)d_in[7];
  const float* je_b2   = (const float*)d_in[8];
  const float* df_bn_g = (const float*)d_in[9];
  const float* df_bn_b = (const float*)d_in[10];
  const float* df_w1   = (const float*)d_in[11];
  const float* df_b1   = (const float*)d_in[12];
  const float* df_w2   = (const float*)d_in[13];
  const float* df_b2   = (const float*)d_in[14];
  const float* sp_w1   = (const float*)d_in[15];
  const float* sp_b1   = (const float*)d_in[16];
  const float* sp_w2   = (const float*)d_in[17];
  const float* sp_b2   = (const float*)d_in[18];
  const float* g1_w    = (const float*)d_in[19];
  const float* g1_b    = (const float*)d_in[20];
  const float* g2_w    = (const float*)d_in[21];
  const float* g2_b    = (const float*)d_in[22];
  const float* gw      = (const float*)d_in[23];
  const float* gw1     = (const float*)d_in[24];
  const float* gw1_b   = (const float*)d_in[25];
  const float* gbn_g   = (const float*)d_in[26];
  const float* gbn_b   = (const float*)d_in[27];

  float* ws      = (float*)d_ws;
  float* inp     = ws + OFF_INP;
  float* bnscale = ws + OFF_BNS;
  float* bnshift = ws + OFF_BNB;
  float* spa_emb = ws + OFF_SPA;
  float* hmax    = ws + OFF_HMAX;
  float* hmin    = ws + OFF_HMIN;
  float* psum    = ws + OFF_PSUM;
  float* psq     = ws + OFF_PSQ;
  float* gmean   = ws + OFF_GM;
  float* ginv    = ws + OFF_GI;

  k_bn_embed_stats<<<150, 256, 0, stream>>>(Angle, dif, je_bn_g, je_bn_b,
                                            df_bn_g, df_bn_b, bnscale, bnshift);
  k_spa_embed<<<1, 32, 0, stream>>>(spa, sp_w1, sp_b1, sp_w2, sp_b2, spa_emb);
  k_embed_gemm<<<12800, 256, 0, stream>>>(Angle, dif, je_w1, je_b1, je_w2, je_b2,
                                          df_w1, df_b1, df_w2, df_b2,
                                          bnscale, bnshift, spa_emb, inp);
  k_attn<<<NBT, 256, 13312 * sizeof(float), stream>>>(inp, g1_w, g1_b, g2_w, g2_b,
                                                      gw, gw1, gw1_b,
                                                      hmax, hmin, psum, psq);
  k_bn2_stats<<<DIMC, 256, 0, stream>>>(psum, psq, gmean, ginv);
  k_finalize<<<NBT, 256, 0, stream>>>(hmax, hmin, gmean, ginv, gbn_g, gbn_b, (float*)d_out);
}